// TreeRejectionSampler_84069689851904
// MI455X (gfx1250) — compile-verified
//
#include <hip/hip_runtime.h>
#include <hip/hip_bf16.h>
#include <float.h>
#include <stdint.h>

// Problem constants (fixed by the reference).
#define BATCH           32
#define TOKENS_PER_TREE 29
#define NUM_INTERNAL    13
#define DRAFT_TREE_SIZE 28
#define VOCAB           128000

#define ARGMAX_BLOCK 256
#define CHUNK (ARGMAX_BLOCK * 4)            // 1024 floats / 4 KB per block-chunk
#define ROW_SPLIT  5                        // partials per 512KB row
#define PART_CHUNKS (VOCAB / CHUNK / ROW_SPLIT)   // 25 chunks per partial
#define PART_FLOATS (PART_CHUNKS * CHUNK)         // 25600 floats per partial
#define NBUF 8                              // async pipeline depth (32 KB LDS)

#define N_INT_ROWS   (BATCH * NUM_INTERNAL)       // 416
#define N_SLOTS      (N_INT_ROWS + BATCH)         // 448 atomic slots

typedef __attribute__((ext_vector_type(4))) float v4f;
typedef __attribute__((ext_vector_type(4))) int   v4i;

// CDNA5 async global->LDS streaming path (gfx1250); pointer types confirmed
// by the round-1 diagnostic (16-byte int vectors in AS1/AS3).
#if defined(__HIP_DEVICE_COMPILE__) &&                                  \
    __has_builtin(__builtin_amdgcn_global_load_async_to_lds_b128) &&    \
    __has_builtin(__builtin_amdgcn_s_wait_asynccnt)
#define USE_ASYNC_LDS 1
typedef __attribute__((address_space(1))) v4i gv4i;
typedef __attribute__((address_space(3))) v4i lv4i;
#define ASYNC_CPOL 1   /* TH_LOAD_NT: read-once 229MB stream, don't wash L2 */
#else
#define USE_ASYNC_LDS 0
#endif

struct MaxPair { float v; int i; };

// Block-wide first-occurrence argmax over nchunk*CHUNK contiguous floats.
// Per-thread scan is in increasing index order with strict '>', and the LDS
// tree reduction breaks value ties by smaller index => jnp.argmax semantics.
__device__ __forceinline__ MaxPair blockRowArgmax(const float* __restrict__ src,
                                                  int baseIdx, int nchunk) {
  const int t = threadIdx.x;
  float best   = -FLT_MAX;
  int   bestId = 0;

#if USE_ASYNC_LDS
  // NBUF-deep async DMA pipeline into LDS. Each lane stages and later consumes
  // exactly its own 16B, so the only synchronization needed is the per-wave
  // ASYNCcnt (async loads complete in order). Steady state keeps NBUF wave-
  // chunks (NBUF*512B per wave) in flight to cover HBM latency.
  __shared__ __align__(16) float sbuf[NBUF][CHUNK];
#pragma unroll
  for (int p = 0; p < NBUF; ++p) {
    __builtin_amdgcn_global_load_async_to_lds_b128(
        (gv4i*)(src + (size_t)p * CHUNK + t * 4),
        (lv4i*)&sbuf[p][t * 4], 0, ASYNC_CPOL);
  }
  for (int c = 0; c < nchunk; ++c) {
    // Oldest outstanding chunk is c; inflight = min(nchunk - c, NBUF).
    // Wait until ASYNCcnt <= inflight-1 (immediate operand -> literal switch).
    const int inflight = (nchunk - c < NBUF) ? (nchunk - c) : NBUF;
    switch (inflight - 1) {
      case 0:  __builtin_amdgcn_s_wait_asynccnt(0); break;
      case 1:  __builtin_amdgcn_s_wait_asynccnt(1); break;
      case 2:  __builtin_amdgcn_s_wait_asynccnt(2); break;
      case 3:  __builtin_amdgcn_s_wait_asynccnt(3); break;
      case 4:  __builtin_amdgcn_s_wait_asynccnt(4); break;
      case 5:  __builtin_amdgcn_s_wait_asynccnt(5); break;
      case 6:  __builtin_amdgcn_s_wait_asynccnt(6); break;
      default: __builtin_amdgcn_s_wait_asynccnt(NBUF - 1); break;
    }
    const v4f v = *(const v4f*)&sbuf[c % NBUF][t * 4];
    const int gi = baseIdx + c * CHUNK + t * 4;
    if (v.x > best) { best = v.x; bestId = gi;     }
    if (v.y > best) { best = v.y; bestId = gi + 1; }
    if (v.z > best) { best = v.z; bestId = gi + 2; }
    if (v.w > best) { best = v.w; bestId = gi + 3; }
    if (c + NBUF < nchunk) {
      // The ds_load above already completed (its data was consumed by the
      // compares) before this overwrite of the same LDS buffer issues.
      __builtin_amdgcn_global_load_async_to_lds_b128(
          (gv4i*)(src + (size_t)(c + NBUF) * CHUNK + t * 4),
          (lv4i*)&sbuf[c % NBUF][t * 4], 0, ASYNC_CPOL);
    }
  }
#else
  for (int c = 0; c < nchunk; ++c) {
    if (c + 8 < nchunk)
      __builtin_prefetch(src + (size_t)(c + 8) * CHUNK + t * 4, 0, 0);
    const v4f v = __builtin_nontemporal_load(
        (const v4f*)(src + (size_t)c * CHUNK + t * 4));
    const int gi = baseIdx + c * CHUNK + t * 4;
    if (v.x > best) { best = v.x; bestId = gi;     }
    if (v.y > best) { best = v.y; bestId = gi + 1; }
    if (v.z > best) { best = v.z; bestId = gi + 2; }
    if (v.w > best) { best = v.w; bestId = gi + 3; }
  }
#endif

  __shared__ float rv[ARGMAX_BLOCK];
  __shared__ int   ri[ARGMAX_BLOCK];
  rv[t] = best; ri[t] = bestId;
  __syncthreads();
  for (int s = ARGMAX_BLOCK / 2; s > 0; s >>= 1) {
    if (t < s) {
      const float ov = rv[t + s]; const int oi = ri[t + s];
      if (ov > rv[t] || (ov == rv[t] && oi < ri[t])) { rv[t] = ov; ri[t] = oi; }
    }
    __syncthreads();
  }
  MaxPair r; r.v = rv[0]; r.i = ri[0];
  __syncthreads();
  return r;
}

// Order-preserving (value, first-index) -> u64 for atomicMax merging.
__device__ __forceinline__ unsigned long long packMax(float v, int idx) {
  unsigned key = __float_as_uint(v);
  key = (key & 0x80000000u) ? ~key : (key | 0x80000000u);
  return ((unsigned long long)key << 32) |
         (unsigned long long)(0xFFFFFFFFu - (unsigned)idx);
}
__device__ __forceinline__ int unpackIdx(unsigned long long p) {
  return (int)(0xFFFFFFFFu - (unsigned)(p & 0xFFFFFFFFull));
}

// ---- Kernel 0: zero the atomic merge slots (graph-replay safe) ----
__global__ __launch_bounds__(512)
void initSlotsKernel(unsigned long long* __restrict__ slots) {
  const int t = threadIdx.x;
  if (t < N_SLOTS) slots[t] = 0ull;
}

// ---- Kernel 1: internal-row argmax partials (213 MB stream, 2080 blocks) ----
__global__ __launch_bounds__(ARGMAX_BLOCK)
void internalPartialKernel(const float* __restrict__ logits,
                           unsigned long long* __restrict__ packedInt) {
  const int row  = blockIdx.x / ROW_SPLIT;    // b*13 + tok
  const int part = blockIdx.x % ROW_SPLIT;
  const int b    = row / NUM_INTERNAL;
  const int tok  = row % NUM_INTERNAL;
  const int partBase = part * PART_FLOATS;
  const float* src =
      logits + ((size_t)b * TOKENS_PER_TREE + tok) * VOCAB + partBase;
  const MaxPair m = blockRowArgmax(src, partBase, PART_CHUNKS);
  if (threadIdx.x == 0) atomicMax(packedInt + row, packMax(m.v, m.i));
}

// Fixed-tree parent function (tokens 0..28).
__device__ __forceinline__ int parentOf(int tok) {
  if (tok <= 0)  return -1;
  if (tok <= 4)  return 0;
  if (tok <= 12) return 1 + (tok - 5) / 2;
  return 5 + (tok - 13) / 2;
}

// ---- Kernel 2: tree acceptance walk, path selection, masked token output ----
// One wave32 per batch. Tree constants are compile-time.
__global__ __launch_bounds__(32)
void treeSelectKernel(const unsigned long long* __restrict__ packedInt,
                      const int* __restrict__ draftIds,
                      int* __restrict__ pathArr,
                      float* __restrict__ out) {
  const int b = blockIdx.x;
  const int t = threadIdx.x;
  const unsigned long long* pm = packedInt + b * NUM_INTERNAL;
  const int* ids = draftIds + b * DRAFT_TREE_SIZE;

  __shared__ int am[NUM_INTERNAL];
  __shared__ int plen[16];
  __shared__ int sPath;

  if (t < NUM_INTERNAL) am[t] = unpackIdx(pm[t]);
  __syncthreads();

  if (t < 16) {
    // path width t: depth-1 token = 1+t/4, depth-2 = 5+t/2, depth-3 = 13+t
    const int a0 = (am[0]            == ids[t >> 2])       ? 1 : 0;
    const int a1 = (am[1 + (t >> 2)] == ids[4 + (t >> 1)]) ? 1 : 0;
    const int a2 = (am[5 + (t >> 1)] == ids[12 + t])       ? 1 : 0;
    plen[t] = a0 ? (a1 ? (a2 ? 3 : 2) : 1) : 0;     // argmin over levels
  }
  __syncthreads();

  if (t == 0) {
    int lvl = 0, idx = 0;
    for (int w = 0; w < 16; ++w)
      if (plen[w] > lvl) { lvl = plen[w]; idx = w; } // first-max (strict >)
    int path;
    if      (lvl == 0) path = 0;
    else if (lvl == 1) path = 1 + (idx >> 2);
    else if (lvl == 2) path = 5 + (idx >> 1);
    else               path = 13 + idx;
    sPath = path;
    pathArr[b] = path;
  }
  __syncthreads();
  const int path = sPath;

  if (t < DRAFT_TREE_SIZE) {
    const int tok = t + 1;
    bool m = false;
    int cur = path;
    while (cur >= 0) { if (cur == tok) m = true; cur = parentOf(cur); }
    out[b * TOKENS_PER_TREE + t] = m ? (float)ids[t] : -1.0f;                 // out28
    out[BATCH * TOKENS_PER_TREE + b * DRAFT_TREE_SIZE + t] = m ? 1.0f : 0.0f; // mask
  }
}

// ---- Kernel 3: bonus-row argmax partials (16 MB stream, 160 blocks) ----
__global__ __launch_bounds__(ARGMAX_BLOCK)
void bonusPartialKernel(const float* __restrict__ logits,
                        const int* __restrict__ pathArr,
                        unsigned long long* __restrict__ packedBonus) {
  const int b    = blockIdx.x / ROW_SPLIT;
  const int part = blockIdx.x % ROW_SPLIT;
  const int path = pathArr[b];
  const int partBase = part * PART_FLOATS;
  const float* src =
      logits + ((size_t)b * TOKENS_PER_TREE + path) * VOCAB + partBase;
  const MaxPair m = blockRowArgmax(src, partBase, PART_CHUNKS);
  if (threadIdx.x == 0) atomicMax(packedBonus + b, packMax(m.v, m.i));
}

// ---- Kernel 4: decode bonus tokens ----
__global__ __launch_bounds__(32)
void bonusDecodeKernel(const unsigned long long* __restrict__ packedBonus,
                       float* __restrict__ out) {
  const int b = threadIdx.x;
  if (b < BATCH) {
    out[b * TOKENS_PER_TREE + DRAFT_TREE_SIZE] = (float)unpackIdx(packedBonus[b]);
  }
}

extern "C" void kernel_launch(void* const* d_in, const int* in_sizes, int n_in,
                              void* d_out, int out_size, void* d_ws, size_t ws_size,
                              hipStream_t stream) {
  const float* logits   = (const float*)d_in[0];
  // d_in[1] temperature: unused — argmax(softmax(x/T)) == argmax(x) for T>0.
  const int*   draftIds = (const int*)d_in[2];
  // d_in[3] tree_mask / d_in[4] tree_draft_positions: recomputed in closed form.

  unsigned long long* packedInt   = (unsigned long long*)d_ws;   // 416 u64
  unsigned long long* packedBonus = packedInt + N_INT_ROWS;      // 32 u64
  int* pathArr = (int*)(packedBonus + BATCH);                    // 32 ints

  initSlotsKernel<<<1, 512, 0, stream>>>(packedInt);             // zero 448 slots
  internalPartialKernel<<<N_INT_ROWS * ROW_SPLIT, ARGMAX_BLOCK, 0, stream>>>(
      logits, packedInt);
  treeSelectKernel<<<BATCH, 32, 0, stream>>>(packedInt, draftIds, pathArr,
                                             (float*)d_out);
  bonusPartialKernel<<<BATCH * ROW_SPLIT, ARGMAX_BLOCK, 0, stream>>>(
      logits, pathArr, packedBonus);
  bonusDecodeKernel<<<1, 32, 0, stream>>>(packedBonus, (float*)d_out);
}